// GNN_29137058136722
// MI455X (gfx1250) — compile-verified
//
#include <hip/hip_runtime.h>

#define NNODES  100000
#define NEDGES  1600000
#define DFEAT   64
#define NLAYERS 5

typedef __attribute__((ext_vector_type(2))) float v2f;
typedef __attribute__((ext_vector_type(8))) float v8f;

// ---------------------------------------------------------------------------
// Degree / normalization kernels
// ---------------------------------------------------------------------------
__global__ void deg_init(float* __restrict__ deg) {
    int i = blockIdx.x * blockDim.x + threadIdx.x;
    if (i < NNODES) deg[i] = 1.0f;               // self-loop contributes 1
}

__global__ void deg_count(const int* __restrict__ dst, float* __restrict__ deg) {
    int e = blockIdx.x * blockDim.x + threadIdx.x;
    if (e < NEDGES) atomicAdd(&deg[dst[e]], 1.0f);
}

__global__ void deg_rsqrt(float* __restrict__ dinv) {
    int i = blockIdx.x * blockDim.x + threadIdx.x;
    if (i < NNODES) dinv[i] = rsqrtf(dinv[i]);   // deg >= 1 always
}

// ---------------------------------------------------------------------------
// Fused WMMA fp32 GEMM + bias + self-loop:
//   XW[N,64]  = X[N,64] @ W[64,64]
//   ACC[N,64] = bias + XW * dinv^2
// One wave32 per 16 rows x all 64 cols: 4 accumulators, A fragment reused
// across 4 v_wmma_f32_16x16x4_f32 per K-step (16 steps, K=64) -> 64 WMMAs.
// A 16x4 layout: lanes 0-15 -> K={0,1}, lanes 16-31 -> K={2,3} (ISA 7.12.2).
// B 4x16 layout: VGPR0 -> K=0|K=2, VGPR1 -> K=1|K=3 striped across lanes.
// C/D 16x16: VGPR v, lanes 0-15 -> M=v, lanes 16-31 -> M=v+8, N=lane&15.
// ---------------------------------------------------------------------------
__global__ __launch_bounds__(32) void gcn_gemm_wmma(const float* __restrict__ X,
                                                    const float* __restrict__ W,
                                                    const float* __restrict__ bias,
                                                    const float* __restrict__ dinv,
                                                    float* __restrict__ XW,
                                                    float* __restrict__ ACC) {
    const int lane = threadIdx.x;                 // 0..31, full wave (EXEC all-1s)
    const int row0 = blockIdx.x * 16;             // M tile base
    const int l15  = lane & 15;
    const int mrow = row0 + l15;
    const int koff = (lane >> 4) * 2;             // 0 for lanes 0-15, 2 for 16-31

    // Per-lane base pointers; everything else is immediate offsets.
    const float* xrow  = X + (size_t)mrow * DFEAT + koff;
    const float* wbase = W + (size_t)koff * DFEAT + l15;

    v8f c0 = {}, c1 = {}, c2 = {}, c3 = {};

#pragma unroll
    for (int ks = 0; ks < DFEAT; ks += 4) {
        v2f a;
        a[0] = xrow[ks];                          // K = ks+koff
        a[1] = xrow[ks + 1];                      // K = ks+koff+1
        v2f b0, b1, b2, b3;
        b0[0] = wbase[ks * DFEAT +  0];  b0[1] = wbase[ks * DFEAT + DFEAT +  0];
        b1[0] = wbase[ks * DFEAT + 16];  b1[1] = wbase[ks * DFEAT + DFEAT + 16];
        b2[0] = wbase[ks * DFEAT + 32];  b2[1] = wbase[ks * DFEAT + DFEAT + 32];
        b3[0] = wbase[ks * DFEAT + 48];  b3[1] = wbase[ks * DFEAT + DFEAT + 48];
        // (neg_a, A, neg_b, B, c_mod, C, reuse_a, reuse_b)
        c0 = __builtin_amdgcn_wmma_f32_16x16x4_f32(false, a, false, b0, (short)0, c0, false, false);
        c1 = __builtin_amdgcn_wmma_f32_16x16x4_f32(false, a, false, b1, (short)0, c1, false, false);
        c2 = __builtin_amdgcn_wmma_f32_16x16x4_f32(false, a, false, b2, (short)0, c2, false, false);
        c3 = __builtin_amdgcn_wmma_f32_16x16x4_f32(false, a, false, b3, (short)0, c3, false, false);
    }

    // Epilogue: write XW and ACC = bias + XW*dinv^2 directly from registers.
    float bv0 = bias[l15 +  0];
    float bv1 = bias[l15 + 16];
    float bv2 = bias[l15 + 32];
    float bv3 = bias[l15 + 48];

    const int mbase = row0 + ((lane >> 4) << 3);  // +8 for upper half-wave
#pragma unroll
    for (int v = 0; v < 8; ++v) {
        const int r = mbase + v;
        const float di = dinv[r];
        const float d2 = di * di;
        const size_t base = (size_t)r * DFEAT + l15;
        float x0 = c0[v], x1 = c1[v], x2 = c2[v], x3 = c3[v];
        XW[base +  0] = x0;  ACC[base +  0] = bv0 + x0 * d2;
        XW[base + 16] = x1;  ACC[base + 16] = bv1 + x1 * d2;
        XW[base + 32] = x2;  ACC[base + 32] = bv2 + x2 * d2;
        XW[base + 48] = x3;  ACC[base + 48] = bv3 + x3 * d2;
    }
}

// ---------------------------------------------------------------------------
// Edge scatter: acc[dst] += xw[src] * dinv[src] * dinv[dst]
// 64 consecutive threads handle one edge -> coalesced 256B gather + atomics.
// ---------------------------------------------------------------------------
__global__ void scatter_edges(const float* __restrict__ xw,
                              const int* __restrict__ src,
                              const int* __restrict__ dst,
                              const float* __restrict__ dinv,
                              float* __restrict__ acc) {
    unsigned int tid = blockIdx.x * blockDim.x + threadIdx.x;
    unsigned int e = tid >> 6;
    if (e >= NEDGES) return;
    unsigned int d = tid & 63;
    int s = src[e];
    int t = dst[e];
    float w = dinv[s] * dinv[t];
    atomicAdd(&acc[(size_t)t * DFEAT + d], xw[(size_t)s * DFEAT + d] * w);
}

// ---------------------------------------------------------------------------
// x_cur = relu(acc + x_orig)
// ---------------------------------------------------------------------------
__global__ void relu_residual(const float* __restrict__ acc,
                              const float* __restrict__ xorig,
                              float* __restrict__ xcur) {
    int idx = blockIdx.x * blockDim.x + threadIdx.x;
    if (idx >= NNODES * DFEAT) return;
    float v = acc[idx] + xorig[idx];
    xcur[idx] = v > 0.0f ? v : 0.0f;
}

// ---------------------------------------------------------------------------
extern "C" void kernel_launch(void* const* d_in, const int* in_sizes, int n_in,
                              void* d_out, int out_size, void* d_ws, size_t ws_size,
                              hipStream_t stream) {
    const float* x_in = (const float*)d_in[0];          // [N, 64]
    const int*   edge = (const int*)d_in[1];            // [2, E]
    const int*   src  = edge;
    const int*   dst  = edge + NEDGES;
    const float* Ws   = (const float*)d_in[2];          // [L, 64, 64]
    const float* bs   = (const float*)d_in[3];          // [L, 64]
    float*       out  = (float*)d_out;                  // [N, 64]

    const size_t feat_bytes = (size_t)NNODES * DFEAT * sizeof(float);
    char* ws = (char*)d_ws;
    float* x_cur   = (float*)(ws);
    float* xw      = (float*)(ws + feat_bytes);
    float* acc_buf = (float*)(ws + 2 * feat_bytes);
    float* dinv    = (float*)(ws + 3 * feat_bytes);

    // --- normalization: deg -> dinv (recomputed every call, stateless) ---
    deg_init <<<(NNODES + 255) / 256, 256, 0, stream>>>(dinv);
    deg_count<<<(NEDGES + 255) / 256, 256, 0, stream>>>(dst, dinv);
    deg_rsqrt<<<(NNODES + 255) / 256, 256, 0, stream>>>(dinv);

    const int gemm_blocks = NNODES / 16;                // 6250 waves
    const int nd_blocks = (NNODES * DFEAT + 255) / 256;
    const unsigned int sc_blocks =
        (unsigned int)(((size_t)NEDGES * DFEAT + 255) / 256);

    for (int l = 0; l < NLAYERS; ++l) {
        const float* xsrc = (l == 0) ? x_in : x_cur;
        float* acc = (l == NLAYERS - 1) ? out : acc_buf;

        gcn_gemm_wmma<<<gemm_blocks, 32, 0, stream>>>(
            xsrc, Ws + (size_t)l * DFEAT * DFEAT, bs + (size_t)l * DFEAT,
            dinv, xw, acc);
        scatter_edges<<<sc_blocks, 256, 0, stream>>>(xw, src, dst, dinv, acc);
        if (l < NLAYERS - 1)
            relu_residual<<<nd_blocks, 256, 0, stream>>>(acc, x_in, x_cur);
    }
}